// RecurrentGCN_33139967656316
// MI455X (gfx1250) — compile-verified
//
#include <hip/hip_runtime.h>
#include <math.h>

// EvolveGCN-O forward: N=100000 nodes, E=1e6 edges, D=128, all fp32.
// Pipeline: gru(W) -> deg -> dinv -> xw=X@W (WMMA f32) -> edge scatter
//           -> out = relu(h + xw*dinv^2) @ lin_w.T + b (WMMA f32, fused self-loop)

typedef float v2f __attribute__((ext_vector_type(2)));
typedef float v8f __attribute__((ext_vector_type(8)));

#define D 128
#define LDS_STRIDE 132   // 132 % 64 = 4 -> conflict-free column reads

__device__ __forceinline__ float sigmoidf_(float x) {
    return 1.0f / (1.0f + __expf(-x));
}

// ---------------------------------------------------------------------------
// 1) GRU step on the weight matrix: W = GRU(x=W0, h=W0). One thread per (m,c).
// ---------------------------------------------------------------------------
__global__ __launch_bounds__(256) void k_gru(const float* __restrict__ W0,
                                             const float* __restrict__ wih,
                                             const float* __restrict__ whh,
                                             const float* __restrict__ bih,
                                             const float* __restrict__ bhh,
                                             float* __restrict__ Wout) {
    int idx = blockIdx.x * 256 + threadIdx.x;   // 128*128 threads
    if (idx >= D * D) return;
    int m = idx >> 7;
    int c = idx & (D - 1);

    const float* xr = W0 + (size_t)m * D;
    float ir = 0.f, iz = 0.f, in_ = 0.f, hr = 0.f, hz = 0.f, hn = 0.f;
    const float* wr_i = wih + (size_t)c * D;
    const float* wz_i = wih + (size_t)(c + D) * D;
    const float* wn_i = wih + (size_t)(c + 2 * D) * D;
    const float* wr_h = whh + (size_t)c * D;
    const float* wz_h = whh + (size_t)(c + D) * D;
    const float* wn_h = whh + (size_t)(c + 2 * D) * D;
#pragma unroll 4
    for (int k = 0; k < D; ++k) {
        float x = xr[k];
        ir += x * wr_i[k];  iz += x * wz_i[k];  in_ += x * wn_i[k];
        hr += x * wr_h[k];  hz += x * wz_h[k];  hn  += x * wn_h[k];
    }
    ir += bih[c];          hr += bhh[c];
    iz += bih[c + D];      hz += bhh[c + D];
    in_ += bih[c + 2 * D]; hn += bhh[c + 2 * D];

    float r = sigmoidf_(ir + hr);
    float z = sigmoidf_(iz + hz);
    float n = tanhf(in_ + r * hn);
    Wout[idx] = (1.0f - z) * n + z * W0[idx];
}

// ---------------------------------------------------------------------------
// 2) Zero h (in d_out) and deg.
// ---------------------------------------------------------------------------
__global__ __launch_bounds__(256) void k_zero(float* __restrict__ h,
                                              float* __restrict__ deg, int N) {
    int gid = blockIdx.x * 256 + threadIdx.x;       // N*32 threads
    if (gid < N * 32) {
        float4 z = make_float4(0.f, 0.f, 0.f, 0.f);
        *(float4*)(h + (size_t)gid * 4) = z;
    }
    if (gid < N) deg[gid] = 0.f;
}

// ---------------------------------------------------------------------------
// 3) In-degree (edges only; self-loop folded into dinv).
// ---------------------------------------------------------------------------
__global__ __launch_bounds__(256) void k_deg(const int* __restrict__ dst,
                                             float* __restrict__ deg, int E) {
    int e = blockIdx.x * 256 + threadIdx.x;
    if (e < E) atomicAdd(&deg[dst[e]], 1.0f);
}

__global__ __launch_bounds__(256) void k_dinv(const float* __restrict__ deg,
                                              float* __restrict__ dinv, int N) {
    int i = blockIdx.x * 256 + threadIdx.x;
    if (i < N) dinv[i] = __frsqrt_rn(deg[i] + 1.0f);   // +1 self-loop, always > 0
}

// ---------------------------------------------------------------------------
// 4) xw = X @ W.  One block = 16 rows x 128 cols; 8 waves, one 16x16 tile each.
//    V_WMMA_F32_16X16X4_F32, 32 steps over K=128.
// ---------------------------------------------------------------------------
__global__ __launch_bounds__(256) void k_gemm_xw(const float* __restrict__ X,
                                                 const float* __restrict__ W,
                                                 float* __restrict__ XW) {
    __shared__ float As[16][LDS_STRIDE];
    const int row0 = blockIdx.x * 16;
    const int tid = threadIdx.x;

    // Stage 16x128 A tile (float4 x 512, 2 per thread).
    for (int i = tid; i < 512; i += 256) {
        int r = i >> 5;
        int c4 = (i & 31) << 2;
        float4 v = *(const float4*)(X + (size_t)(row0 + r) * D + c4);
        As[r][c4 + 0] = v.x; As[r][c4 + 1] = v.y;
        As[r][c4 + 2] = v.z; As[r][c4 + 3] = v.w;
    }
    __syncthreads();

    const int lane = tid & 31;
    const int mn   = lane & 15;          // A: row M / B: col N
    const int koff = (lane >> 4) << 1;   // K sub-offset {0,2}
    const int col0 = (tid >> 5) << 4;    // wave -> 16-col tile

    v8f c = {0.f, 0.f, 0.f, 0.f, 0.f, 0.f, 0.f, 0.f};
#pragma unroll
    for (int k0 = 0; k0 < D; k0 += 4) {
        v2f a = *(const v2f*)&As[mn][k0 + koff];            // A[M][k], A[M][k+1]
        v2f b;
        b.x = W[(size_t)(k0 + koff) * D + col0 + mn];       // B[k][N]
        b.y = W[(size_t)(k0 + koff + 1) * D + col0 + mn];   // B[k+1][N]
        c = __builtin_amdgcn_wmma_f32_16x16x4_f32(false, a, false, b,
                                                  (short)0, c, false, false);
    }

    const int rbase = row0 + ((lane >> 4) << 3);   // M = vgpr + {0,8}
    const int col = col0 + mn;
#pragma unroll
    for (int r = 0; r < 8; ++r)
        XW[(size_t)(rbase + r) * D + col] = c[r];
}

// ---------------------------------------------------------------------------
// 5) Edge scatter: one wave per edge, float4 per lane (32*4 = 128 floats).
//    h[dst] += xw[src] * dinv[src]*dinv[dst]   via f32 atomics.
// ---------------------------------------------------------------------------
__global__ __launch_bounds__(256) void k_edges(const int* __restrict__ src,
                                               const int* __restrict__ dst,
                                               const float* __restrict__ xw,
                                               const float* __restrict__ dinv,
                                               float* __restrict__ h, int E) {
    int gid = blockIdx.x * 256 + threadIdx.x;
    int e = gid >> 5;
    if (e >= E) return;
    int lane = gid & 31;
    int s = src[e];
    int d = dst[e];
    float nrm = dinv[s] * dinv[d];
    int cofs = lane << 2;
    float4 v = *(const float4*)(xw + (size_t)s * D + cofs);
    float* hp = h + (size_t)d * D + cofs;
    atomicAdd(hp + 0, v.x * nrm);
    atomicAdd(hp + 1, v.y * nrm);
    atomicAdd(hp + 2, v.z * nrm);
    atomicAdd(hp + 3, v.w * nrm);
}

// ---------------------------------------------------------------------------
// 6) out = relu(h + xw*dinv^2) @ lin_w.T + lin_b  (WMMA, self-loop fused).
//    h is d_out; output overwrites the same 16 rows this block staged to LDS.
// ---------------------------------------------------------------------------
__global__ __launch_bounds__(256) void k_gemm_out(const float* __restrict__ XW,
                                                  const float* __restrict__ dinv,
                                                  const float* __restrict__ Lw,
                                                  const float* __restrict__ Lb,
                                                  float* __restrict__ HO) {
    __shared__ float As[16][LDS_STRIDE];
    const int row0 = blockIdx.x * 16;
    const int tid = threadIdx.x;

    for (int i = tid; i < 512; i += 256) {
        int r = i >> 5;
        int c4 = (i & 31) << 2;
        size_t base = (size_t)(row0 + r) * D + c4;
        float dv = dinv[row0 + r];
        float d2 = dv * dv;
        float4 hv = *(const float4*)(HO + base);
        float4 xv = *(const float4*)(XW + base);
        As[r][c4 + 0] = fmaxf(hv.x + xv.x * d2, 0.f);
        As[r][c4 + 1] = fmaxf(hv.y + xv.y * d2, 0.f);
        As[r][c4 + 2] = fmaxf(hv.z + xv.z * d2, 0.f);
        As[r][c4 + 3] = fmaxf(hv.w + xv.w * d2, 0.f);
    }
    __syncthreads();

    const int lane = tid & 31;
    const int mn   = lane & 15;
    const int koff = (lane >> 4) << 1;
    const int col0 = (tid >> 5) << 4;

    v8f c = {0.f, 0.f, 0.f, 0.f, 0.f, 0.f, 0.f, 0.f};
#pragma unroll
    for (int k0 = 0; k0 < D; k0 += 4) {
        v2f a = *(const v2f*)&As[mn][k0 + koff];
        v2f b;  // B[k][j] = lin_w[j][k]  (lin_w.T)
        b.x = Lw[(size_t)(col0 + mn) * D + (k0 + koff)];
        b.y = Lw[(size_t)(col0 + mn) * D + (k0 + koff + 1)];
        c = __builtin_amdgcn_wmma_f32_16x16x4_f32(false, a, false, b,
                                                  (short)0, c, false, false);
    }

    const int rbase = row0 + ((lane >> 4) << 3);
    const int col = col0 + mn;
    const float bias = Lb[col];
#pragma unroll
    for (int r = 0; r < 8; ++r)
        HO[(size_t)(rbase + r) * D + col] = c[r] + bias;
}

// ---------------------------------------------------------------------------
extern "C" void kernel_launch(void* const* d_in, const int* in_sizes, int n_in,
                              void* d_out, int out_size, void* d_ws, size_t ws_size,
                              hipStream_t stream) {
    const float* node_feat = (const float*)d_in[0];
    const int*   src       = (const int*)d_in[1];
    const int*   dst       = (const int*)d_in[2];
    const float* W0        = (const float*)d_in[3];
    const float* gru_w_ih  = (const float*)d_in[4];
    const float* gru_w_hh  = (const float*)d_in[5];
    const float* gru_b_ih  = (const float*)d_in[6];
    const float* gru_b_hh  = (const float*)d_in[7];
    const float* lin_w     = (const float*)d_in[8];
    const float* lin_b     = (const float*)d_in[9];

    const int N = in_sizes[0] / D;   // 100000 (multiple of 16)
    const int E = in_sizes[1];       // 1000000

    // Workspace layout (512B-aligned blocks).
    char* ws = (char*)d_ws;
    auto align512 = [](size_t x) { return (x + 511) & ~(size_t)511; };
    size_t off = 0;
    float* W    = (float*)(ws + off); off = align512(off + (size_t)D * D * 4);
    float* deg  = (float*)(ws + off); off = align512(off + (size_t)N * 4);
    float* dinv = (float*)(ws + off); off = align512(off + (size_t)N * 4);
    float* xw   = (float*)(ws + off); off = align512(off + (size_t)N * D * 4);
    (void)ws_size;

    float* h = (float*)d_out;   // accumulator; final GEMM overwrites in place

    // 1) evolve weights
    k_gru<<<(D * D + 255) / 256, 256, 0, stream>>>(W0, gru_w_ih, gru_w_hh,
                                                   gru_b_ih, gru_b_hh, W);
    // 2) zero h + deg
    k_zero<<<(N * 32 + 255) / 256, 256, 0, stream>>>(h, deg, N);
    // 3) degree + norm
    k_deg<<<(E + 255) / 256, 256, 0, stream>>>(dst, deg, E);
    k_dinv<<<(N + 255) / 256, 256, 0, stream>>>(deg, dinv, N);
    // 4) xw = X @ W
    k_gemm_xw<<<N / 16, 256, 0, stream>>>(node_feat, W, xw);
    // 5) edge scatter-add
    k_edges<<<(E * 32 + 255) / 256, 256, 0, stream>>>(src, dst, xw, dinv, h, E);
    // 6) out = relu(h + self-loop) @ lin_w.T + lin_b
    k_gemm_out<<<N / 16, 256, 0, stream>>>(xw, dinv, lin_w, lin_b, h);
}